// ConvLayer_7971459301987
// MI455X (gfx1250) — compile-verified
//
#include <hip/hip_runtime.h>
#include <math.h>

#define DD 64
#define ETA 0.5f
#define NEG_SLOPE 0.01f

typedef __attribute__((ext_vector_type(2))) float v2f;
typedef __attribute__((ext_vector_type(8))) float v8f;

// ---------------- init: m_f,m_t = -inf ; s_f,s_t,z = 0 ----------------
__global__ void init_kernel(float* __restrict__ p, int n_neg_inf, int n_total) {
  int i = blockIdx.x * blockDim.x + threadIdx.x;
  if (i < n_total) p[i] = (i < n_neg_inf) ? -__builtin_inff() : 0.0f;
}

// ---------------- per-node logit halves: a_src = h·wh[0:64], a_dst = h·wh[64:128]
__global__ void node_pre_kernel(const float* __restrict__ h,
                                const float* __restrict__ wh,
                                float* __restrict__ a_src,
                                float* __restrict__ a_dst, int N) {
  int node = (blockIdx.x * blockDim.x + threadIdx.x) >> 5;  // one wave32 per node
  int lane = threadIdx.x & 31;
  if (node >= N) return;
  const float* hp = h + (size_t)node * DD;
  float h0 = hp[lane], h1 = hp[lane + 32];
  float ps = h0 * wh[lane]      + h1 * wh[lane + 32];
  float pd = h0 * wh[64 + lane] + h1 * wh[96 + lane];
#pragma unroll
  for (int off = 16; off > 0; off >>= 1) {
    ps += __shfl_xor(ps, off, 32);
    pd += __shfl_xor(pd, off, 32);
  }
  if (lane == 0) { a_src[node] = ps; a_dst[node] = pd; }
}

// ---------------- edge pass 1: wt dot, wf leaky, atomicMax segment maxima ----
__global__ void edge_logit_kernel(const float* __restrict__ tax,
                                  const int* __restrict__ src,
                                  const int* __restrict__ dst,
                                  const float* __restrict__ a_src,
                                  const float* __restrict__ a_dst,
                                  float* __restrict__ wf, float* __restrict__ wt,
                                  float* __restrict__ m_f, float* __restrict__ m_t,
                                  int E) {
  int t   = blockIdx.x * blockDim.x + threadIdx.x;
  int e   = t >> 3;        // 8 lanes per edge
  int sub = t & 7;
  if (e >= E) return;
  int s = src[e], d = dst[e];
  const float4* ts = (const float4*)(tax + (size_t)s * DD) + sub * 2;
  const float4* td = (const float4*)(tax + (size_t)d * DD) + sub * 2;
  float4 as0 = ts[0], as1 = ts[1];
  float4 ad0 = td[0], ad1 = td[1];
  float dot = as0.x * ad0.x + as0.y * ad0.y + as0.z * ad0.z + as0.w * ad0.w
            + as1.x * ad1.x + as1.y * ad1.y + as1.z * ad1.z + as1.w * ad1.w;
  dot += __shfl_xor(dot, 1, 32);
  dot += __shfl_xor(dot, 2, 32);
  dot += __shfl_xor(dot, 4, 32);
  if (sub == 0) {
    float x  = a_src[s] + a_dst[d];
    float lf = x > 0.0f ? x : NEG_SLOPE * x;
    wf[e] = lf;
    wt[e] = dot;
    atomicMax(m_f + d, lf);
    atomicMax(m_t + d, dot);
  }
}

// ---------------- edge pass 2: exponentiate, accumulate segment sums --------
__global__ void edge_exp_kernel(const int* __restrict__ dst,
                                float* __restrict__ wf, float* __restrict__ wt,
                                const float* __restrict__ m_f,
                                const float* __restrict__ m_t,
                                float* __restrict__ s_f, float* __restrict__ s_t,
                                int E) {
  int e = blockIdx.x * blockDim.x + threadIdx.x;
  if (e >= E) return;
  int d = dst[e];
  float ef = __expf(wf[e] - m_f[d]);
  float et = __expf(wt[e] - m_t[d]);
  wf[e] = ef;
  wt[e] = et;
  atomicAdd(s_f + d, ef);
  atomicAdd(s_t + d, et);
}

// ---------------- edge pass 3: z[dst] += alpha * h[src] ---------------------
__global__ void edge_agg_kernel(const float* __restrict__ h,
                                const int* __restrict__ src,
                                const int* __restrict__ dst,
                                const float* __restrict__ wf,
                                const float* __restrict__ wt,
                                const float* __restrict__ s_f,
                                const float* __restrict__ s_t,
                                float* __restrict__ z, int E) {
  int t   = blockIdx.x * blockDim.x + threadIdx.x;
  int e   = t >> 3;
  int sub = t & 7;
  if (e >= E) return;
  int s = src[e], d = dst[e];
  float alpha = ETA * wf[e] / s_f[d] + (1.0f - ETA) * wt[e] / s_t[d];
  const float4* hp = (const float4*)(h + (size_t)s * DD) + sub * 2;
  float4 v0 = hp[0], v1 = hp[1];
  float* zp = z + (size_t)d * DD + sub * 8;
  atomicAdd(zp + 0, alpha * v0.x);
  atomicAdd(zp + 1, alpha * v0.y);
  atomicAdd(zp + 2, alpha * v0.z);
  atomicAdd(zp + 3, alpha * v0.w);
  atomicAdd(zp + 4, alpha * v1.x);
  atomicAdd(zp + 5, alpha * v1.y);
  atomicAdd(zp + 6, alpha * v1.z);
  atomicAdd(zp + 7, alpha * v1.w);
}

// ---------------- out = z @ W^T + b via V_WMMA_F32_16X16X4_F32 --------------
// One wave computes a 16-row x 64-col output block.
// A (16x4 f32): lanes 0-15 hold K=k0,k0+1 ; lanes 16-31 hold K=k0+2,k0+3.
// B (4x16 f32): VGPR0 lanes0-15 K=0 / lanes16-31 K=2 ; VGPR1 K=1 / K=3.
// D (16x16 f32): VGPR r = rows r (lanes 0-15) and r+8 (lanes 16-31).
__global__ void out_gemm_kernel(const float* __restrict__ z,
                                const float* __restrict__ W,
                                const float* __restrict__ bias,
                                float* __restrict__ out, int N) {
  int wave = (blockIdx.x * blockDim.x + threadIdx.x) >> 5;
  int lane = threadIdx.x & 31;
  int m0 = wave * 16;
  if (m0 >= N) return;                       // whole wave exits together (EXEC all-1 for WMMA)
  int half = lane >> 4;                      // 0: lanes 0-15, 1: lanes 16-31
  int l15  = lane & 15;

  v8f acc[4] = {};
  const float* zrow = z + (size_t)(m0 + l15) * DD;

  for (int kb = 0; kb < DD; kb += 4) {       // 16 K-steps of 4
    v2f a;
    a.x = zrow[kb + half * 2];
    a.y = zrow[kb + half * 2 + 1];
#pragma unroll
    for (int tcol = 0; tcol < 4; ++tcol) {   // 4 column tiles of 16
      int col = tcol * 16 + l15;
      v2f b;
      // B[k][i] = W[i][kb+k]  (out[n][i] = sum_j z[n][j] * W[i][j])
      b.x = W[col * DD + kb + half * 2];
      b.y = W[col * DD + kb + half * 2 + 1];
      acc[tcol] = __builtin_amdgcn_wmma_f32_16x16x4_f32(
          /*neg_a=*/false, a, /*neg_b=*/false, b,
          /*c_mod=*/(short)0, acc[tcol], /*reuse_a=*/false, /*reuse_b=*/false);
    }
  }

#pragma unroll
  for (int tcol = 0; tcol < 4; ++tcol) {
    int col = tcol * 16 + l15;
    float bv = bias[col];
#pragma unroll
    for (int r = 0; r < 8; ++r) {
      int row = m0 + r + half * 8;
      out[(size_t)row * DD + col] = acc[tcol][r] + bv;
    }
  }
}

extern "C" void kernel_launch(void* const* d_in, const int* in_sizes, int n_in,
                              void* d_out, int out_size, void* d_ws, size_t ws_size,
                              hipStream_t stream) {
  const float* h   = (const float*)d_in[0];
  const float* tax = (const float*)d_in[1];
  const int*   src = (const int*)d_in[2];
  const int*   dst = (const int*)d_in[3];
  const float* wh  = (const float*)d_in[4];
  const float* W   = (const float*)d_in[5];
  const float* Wb  = (const float*)d_in[6];
  float* out = (float*)d_out;

  int N = in_sizes[0] / DD;
  int E = in_sizes[2];

  // workspace layout (floats)
  float* ws    = (float*)d_ws;
  float* a_src = ws;                         // N
  float* a_dst = a_src + N;                  // N
  float* m_f   = a_dst + N;                  // N   (-inf init)
  float* m_t   = m_f + N;                    // N   (-inf init)
  float* s_f   = m_t + N;                    // N   (0 init)
  float* s_t   = s_f + N;                    // N   (0 init)
  float* z     = s_t + N;                    // N*DD (0 init)
  float* wf    = z + (size_t)N * DD;         // E
  float* wt    = wf + E;                     // E

  // init m_f..z region: first 2N entries -> -inf, rest -> 0
  int initN = 4 * N + N * DD;
  init_kernel<<<(initN + 255) / 256, 256, 0, stream>>>(m_f, 2 * N, initN);

  node_pre_kernel<<<(N + 7) / 8, 256, 0, stream>>>(h, wh, a_src, a_dst, N);

  int lt = E * 8;  // 8 lanes per edge
  edge_logit_kernel<<<(lt + 255) / 256, 256, 0, stream>>>(
      tax, src, dst, a_src, a_dst, wf, wt, m_f, m_t, E);

  edge_exp_kernel<<<(E + 255) / 256, 256, 0, stream>>>(
      dst, wf, wt, m_f, m_t, s_f, s_t, E);

  edge_agg_kernel<<<(lt + 255) / 256, 256, 0, stream>>>(
      h, src, dst, wf, wt, s_f, s_t, z, E);

  int tiles = (N + 15) / 16;
  int gth = tiles * 32;
  out_gemm_kernel<<<(gth + 255) / 256, 256, 0, stream>>>(z, W, Wb, out, N);
}